// ChannelAttention_72584947302498
// MI455X (gfx1250) — compile-verified
//
#include <hip/hip_runtime.h>

#define N_   64
#define C_   256
#define T_   64
#define V_   25
#define HID_ 16
#define NG_  5

typedef __attribute__((ext_vector_type(2))) float v2f;
typedef __attribute__((ext_vector_type(4))) float v4f;
typedef __attribute__((ext_vector_type(8))) float v8f;

// Output joint order = concat(TORSO, LEFT_HAND, LEFT_LEG, RIGHT_HAND, RIGHT_LEG)
__constant__ int c_JMAP[V_] = {0,1,2,3,20, 8,9,10,11,23,24, 16,17,18,19, 4,5,6,7,21,22, 12,13,14,15};
__constant__ int c_GRP [V_] = {0,0,0,0,0,  1,1,1,1,1,1,     2,2,2,2,     3,3,3,3,3,3,   4,4,4,4};

// ---------------------------------------------------------------------------
// Kernel 1: avg + max pool over torso joints. One wave per (n,c) pair.
// pool layout: [0][n][c] = mean, [1][n][c] = max  (2 * N * C floats)
// Side effect: pulls x's cache lines into the 192 MB L2 ahead of apply_kernel.
// ---------------------------------------------------------------------------
__global__ __launch_bounds__(256) void pool_kernel(const float* __restrict__ x,
                                                   float* __restrict__ pool) {
  const int wid  = threadIdx.x >> 5;
  const int lane = threadIdx.x & 31;
  const int p    = blockIdx.x * 8 + wid;          // (n,c) flat index, 0..N*C-1
  const float* base = x + (size_t)p * (T_ * V_);

  float s = 0.0f;
  float m = -__builtin_inff();
  #pragma unroll
  for (int i = 0; i < 10; ++i) {                  // 320 torso elems / 32 lanes
    int idx = lane + i * 32;                      // 0..319
    int t = idx / 5;
    int r = idx - t * 5;
    int v = (r == 4) ? 20 : r;                    // TORSO = {0,1,2,3,20}
    float val = base[t * V_ + v];
    s += val;
    m  = fmaxf(m, val);
  }
  #pragma unroll
  for (int off = 16; off >= 1; off >>= 1) {
    s += __shfl_xor(s, off, 32);
    m  = fmaxf(m, __shfl_xor(m, off, 32));
  }
  if (lane == 0) {
    pool[p]           = s * (1.0f / 320.0f);      // mean plane
    pool[N_ * C_ + p] = m;                        // max plane
  }
}

// ---------------------------------------------------------------------------
// Kernel 2: gates = sigmoid(mlp(avg) + mlp(mx)) via V_WMMA_F32_16X16X4_F32.
// Grid = 5 groups; block = 128 (4 waves), wave w owns batch rows 16w..16w+15.
// ---------------------------------------------------------------------------
__global__ __launch_bounds__(128) void gates_kernel(const float* __restrict__ pool,
                                                    const float* __restrict__ W1s,
                                                    const float* __restrict__ b1s,
                                                    const float* __restrict__ W2s,
                                                    const float* __restrict__ b2s,
                                                    float* __restrict__ gates) {
  __shared__ float lds_h[4][2][16][17];           // [wave][branch][row][hid] (+pad)

  const int f    = blockIdx.x;                    // group 0..4
  const int wv   = threadIdx.x >> 5;              // M-tile within batch
  const int lane = threadIdx.x & 31;
  const int row  = lane & 15;                     // A: M row / B,C,D: column
  const int kb   = (lane >> 4) << 1;              // K offset 0 or 2 (ISA A layout)
  const int m    = wv * 16;                       // first batch row of this tile

  const float* pa = pool;                         // avg (N x C)
  const float* pm = pool + N_ * C_;               // max (N x C)
  const float* W1 = W1s + f * HID_ * C_;          // (HID, C) row-major
  const float* W2 = W2s + f * C_ * HID_;          // (C, HID) row-major

  // ---- GEMM1: h = relu(P @ W1^T + b1), both branches, shared B operand ----
  v8f ha = {}, hm = {};
  for (int k0 = 0; k0 < C_; k0 += 4) {
    const int k = k0 + kb;
    v2f a0, a1, b;
    a0.x = pa[(m + row) * C_ + k];  a0.y = pa[(m + row) * C_ + k + 1];
    a1.x = pm[(m + row) * C_ + k];  a1.y = pm[(m + row) * C_ + k + 1];
    b.x  = W1[row * C_ + k];        b.y  = W1[row * C_ + k + 1];  // B[k][col]=W1[col][k]
    ha = __builtin_amdgcn_wmma_f32_16x16x4_f32(false, a0, false, b, (short)0, ha, false, false);
    hm = __builtin_amdgcn_wmma_f32_16x16x4_f32(false, a1, false, b, (short)0, hm, false, false);
  }

  // bias + relu, then stage to LDS (C-layout -> row/col scratch)
  const float b1v = b1s[f * HID_ + row];          // col = lane&15
  const int srow  = (lane < 16) ? 0 : 8;
  #pragma unroll
  for (int r = 0; r < 8; ++r) {
    lds_h[wv][0][srow + r][row] = fmaxf(ha[r] + b1v, 0.0f);
    lds_h[wv][1][srow + r][row] = fmaxf(hm[r] + b1v, 0.0f);
  }
  __syncthreads();

  // ---- GEMM2: out = h @ W2^T, both branches into one accumulator ----------
  for (int j = 0; j < 16; ++j) {                  // 16 column tiles of C=256
    v8f acc = {};
    #pragma unroll
    for (int kk = 0; kk < 4; ++kk) {              // K = HID = 16
      const int k = kk * 4 + kb;
      v2f aA, aM, b;
      aA.x = lds_h[wv][0][row][k];  aA.y = lds_h[wv][0][row][k + 1];
      aM.x = lds_h[wv][1][row][k];  aM.y = lds_h[wv][1][row][k + 1];
      b.x  = W2[(j * 16 + row) * HID_ + k];       // B[k][col] = W2[j*16+col][k]
      b.y  = W2[(j * 16 + row) * HID_ + k + 1];
      acc = __builtin_amdgcn_wmma_f32_16x16x4_f32(false, aA, false, b, (short)0, acc, false, false);
      acc = __builtin_amdgcn_wmma_f32_16x16x4_f32(false, aM, false, b, (short)0, acc, false, false);
    }
    const int   cch = j * 16 + row;
    const float b2v = 2.0f * b2s[f * C_ + cch];   // both branches add b2
    #pragma unroll
    for (int r = 0; r < 8; ++r) {
      const int nrow = m + ((lane < 16) ? r : r + 8);
      const float val = acc[r] + b2v;
      const float g   = 1.0f / (1.0f + __expf(-val));
      gates[(nrow * NG_ + f) * C_ + cch] = g;     // gates[n][f][c]
    }
  }
}

// ---------------------------------------------------------------------------
// Kernel 3: out[n,c,t,j] = x[n,c,t,JMAP[j]] * gates[n][GRP[j]][c]
// One block per (n,c). Gathered b32 reads (should hit L2-resident x),
// 16B-aligned b128 NON-TEMPORAL stores so the write stream doesn't evict x.
// ---------------------------------------------------------------------------
__global__ __launch_bounds__(256) void apply_kernel(const float* __restrict__ x,
                                                    const float* __restrict__ gates,
                                                    float* __restrict__ out) {
  const int p = blockIdx.x;                       // (n,c): n = p>>8, c = p&255
  const int n = p >> 8;
  const int c = p & 255;

  float g[NG_];
  #pragma unroll
  for (int k = 0; k < NG_; ++k) g[k] = gates[(n * NG_ + k) * C_ + c];

  const float* xb = x   + (size_t)p * (T_ * V_);
  float*       ob = out + (size_t)p * (T_ * V_);

  // 1600 floats per (n,c) row = 400 aligned float4 groups (row base is 6400B).
  for (int q = threadIdx.x; q < (T_ * V_) / 4; q += 256) {
    const int idx = q * 4;
    int t = idx / V_;
    int j = idx - t * V_;                         // group of 4 may span two t rows
    v4f val;
    #pragma unroll
    for (int e = 0; e < 4; ++e) {
      val[e] = xb[t * V_ + c_JMAP[j]] * g[c_GRP[j]];
      if (++j == V_) { j = 0; ++t; }
    }
    __builtin_nontemporal_store(val, (v4f*)(ob + idx));
  }
}

// ---------------------------------------------------------------------------
extern "C" void kernel_launch(void* const* d_in, const int* in_sizes, int n_in,
                              void* d_out, int out_size, void* d_ws, size_t ws_size,
                              hipStream_t stream) {
  const float* x   = (const float*)d_in[0];   // (N, C, T, V)
  const float* W1s = (const float*)d_in[1];   // (5, HID, C)
  const float* b1s = (const float*)d_in[2];   // (5, HID)
  const float* W2s = (const float*)d_in[3];   // (5, C, HID)
  const float* b2s = (const float*)d_in[4];   // (5, C)
  float* out = (float*)d_out;

  float* pool  = (float*)d_ws;                // 2 * N * C floats
  float* gates = pool + 2 * N_ * C_;          // N * 5 * C floats

  pool_kernel <<<(N_ * C_) / 8, 256, 0, stream>>>(x, pool);
  gates_kernel<<<NG_,          128, 0, stream>>>(pool, W1s, b1s, W2s, b2s, gates);
  apply_kernel<<<N_ * C_,      256, 0, stream>>>(x, gates, out);
}